// CrossAttnQKVModule_13537736917629
// MI455X (gfx1250) — compile-verified
//
#include <hip/hip_runtime.h>

typedef __attribute__((ext_vector_type(2))) float v2f;
typedef __attribute__((ext_vector_type(4))) float v4f;
typedef __attribute__((ext_vector_type(8))) float v8f;

// ---------------- problem constants ----------------
#define NB    2
#define NTOK  4096
#define CMV   32
#define CS    128

// repacked-weight fragment bases in d_ws (float units)
// frag = 32 lanes x 2 floats = 64 floats, A-matrix 16x4 f32 layout
#define WQ_MV     0        // [y:9][ot:4][ks:8]
#define WKV_MV    18432    // [y:9][ot:8][ks:8]
#define WQ_S2MV   55296    // [ot:4][ks:32]
#define WKV_S2MV  63488    // [ot:8][ks:32]
#define WQ_MV2S   79872    // [ot:16][ks:8]
#define WKV_MV2S  88064    // [ot:32][ks:8]
#define WQ_S2S    104448   // [ot:16][ks:32]
#define WKV_S2S   137216   // [ot:32][ks:32]
// total = 202752 floats = 811,008 bytes of workspace

// output offsets (floats), reference tuple order
#define O_QMV   0
#define O_KMV   8388608
#define O_VMV   16777216
#define O_VMVQ  25165824
#define O_QS    33554432
#define O_KS    35651584
#define O_VS    37748736
#define O_VSQ   39845888

// LDS strides (dwords), padded for conflict-free ds_load_b64 B-fragments
#define MV_STRIDE 36   // mv[x][t][i], i-stride
#define S_STRIDE  132  // s[t][i]

// ---- force the global (addrspace 1) memory path
static __device__ __forceinline__ v2f ldg_v2f(const float* p) {
  return *(const v2f __attribute__((address_space(1)))*)p;
}
static __device__ __forceinline__ v4f ldg_v4f(const float* p) {
  return *(const v4f __attribute__((address_space(1)))*)p;
}
static __device__ __forceinline__ float ldg_f(const float* p) {
  return *(const float __attribute__((address_space(1)))*)p;
}
static __device__ __forceinline__ int ldg_i(const int* p) {
  return *(const int __attribute__((address_space(1)))*)p;
}
static __device__ __forceinline__ void stg_f(float* p, float v) {
  *(float __attribute__((address_space(1)))*)p = v;
}
static __device__ __forceinline__ void stg_v2f(float* p, v2f v) {
  *(v2f __attribute__((address_space(1)))*)p = v;
}

static __device__ __forceinline__ v8f wmma_f32(v2f a, v2f b, v8f c) {
  return __builtin_amdgcn_wmma_f32_16x16x4_f32(false, a, false, b, (short)0, c,
                                               false, false);
}

// ---------------- weight repack into WMMA A-fragments ----------------
// frag f = (y*numOT + ot)*numKS + ks ; lane l holds
// W[ot*16 + l%16][ks*4 + (l/16)*2 + {0,1}] (A 16x4 f32 layout)
__global__ void repack_kernel(const float* __restrict__ qwmv,
                              const float* __restrict__ kvwmv,
                              const float* __restrict__ qws2mv,
                              const float* __restrict__ kvws2mv,
                              const float* __restrict__ qwmv2s,
                              const float* __restrict__ kvwmv2s,
                              const float* __restrict__ qws2s,
                              const float* __restrict__ kvws2s,
                              float* __restrict__ ws) {
  int set = blockIdx.y;
  int tid = blockIdx.x * blockDim.x + threadIdx.x;
  int O, I, ny, ys, base;
  const float* src;
  switch (set) {
    case 0:  O = 64;  I = 32;  ny = 9; ys = 9; src = qwmv;    base = WQ_MV;    break;
    case 1:  O = 128; I = 32;  ny = 9; ys = 9; src = kvwmv;   base = WKV_MV;   break;
    case 2:  O = 64;  I = 128; ny = 1; ys = 1; src = qws2mv;  base = WQ_S2MV;  break;
    case 3:  O = 128; I = 128; ny = 1; ys = 1; src = kvws2mv; base = WKV_S2MV; break;
    case 4:  O = 256; I = 32;  ny = 1; ys = 1; src = qwmv2s;  base = WQ_MV2S;  break;
    case 5:  O = 512; I = 32;  ny = 1; ys = 1; src = kvwmv2s; base = WKV_MV2S; break;
    case 6:  O = 256; I = 128; ny = 1; ys = 1; src = qws2s;   base = WQ_S2S;   break;
    default: O = 512; I = 128; ny = 1; ys = 1; src = kvws2s;  base = WKV_S2S;  break;
  }
  int nk = I >> 2, not_ = O >> 4;
  int f = tid >> 5;
  if (f >= ny * not_ * nk) return;
  int lane = tid & 31;
  int ks = f % nk;
  int rest = f / nk;
  int ot = rest % not_;
  int y = rest / not_;
  int o = ot * 16 + (lane & 15);
  int i = ks * 4 + ((lane >> 4) << 1);
  v2f w;
  w.x = ldg_f(src + (o * I + i) * ys + y);
  w.y = ldg_f(src + (o * I + i + 1) * ys + y);
  stg_v2f(ws + base + f * 64 + lane * 2, w);
}

// ---------------- staging: gather 16 tokens into LDS ----------------
static __device__ __forceinline__ void stage_tile(
    const float* __restrict__ inputs, const float* __restrict__ scalars,
    const int* __restrict__ idx, int bb, int tile16, int tid,
    float* __restrict__ mvlds, float* __restrict__ slds) {
  // multivectors: 16 tok * 32 ch * 16 blades, transpose to mv[blade][t][i]
#pragma unroll
  for (int it = 0; it < 8; ++it) {
    int g = it * 256 + tid;        // 0..2047 float4 chunks
    int t = g >> 7;
    int r = g & 127;
    int e = r * 4;                 // element in 512-float token row (i*16+a)
    int i = e >> 4;
    int a0 = e & 15;
    int n = ldg_i(idx + tile16 + t);
    const v4f v4 =
        ldg_v4f(inputs + (((long)bb * NTOK + n) * CMV + i) * 16 + a0);
    mvlds[((a0 + 0) * 16 + t) * MV_STRIDE + i] = v4.x;
    mvlds[((a0 + 1) * 16 + t) * MV_STRIDE + i] = v4.y;
    mvlds[((a0 + 2) * 16 + t) * MV_STRIDE + i] = v4.z;
    mvlds[((a0 + 3) * 16 + t) * MV_STRIDE + i] = v4.w;
  }
  // scalars: 16 tok * 128 -> s[t][i]
#pragma unroll
  for (int it = 0; it < 2; ++it) {
    int g = it * 256 + tid;        // 0..511 float4 chunks
    int t = g >> 5;
    int i0 = (g & 31) * 4;
    int n = ldg_i(idx + tile16 + t);
    *(v4f*)(slds + t * S_STRIDE + i0) =
        ldg_v4f(scalars + ((long)bb * NTOK + n) * CS + i0);
  }
}

// ---------------- one 16-channel multivector output tile ----------------
// Blades grouped by grade so grade/e0 A-fragments are loaded once into VGPRs
// and reused across the blades of the group (cuts A traffic ~2x).
static __device__ void run_mv_job(const float* __restrict__ wsf, int wmvBase,
                                  int s2mvBase, int OT, int ot,
                                  const float* __restrict__ mvlds,
                                  const float* __restrict__ slds,
                                  float* __restrict__ dlo,
                                  float* __restrict__ dhi, int split, int bb,
                                  int tile16, int lane) {
  const int t = lane & 15;
  const int hi = lane >> 4;
  const int n = tile16 + t;
  const int laneoff = lane * 2;

  // tile never straddles the k/v split -> uniform scalar selection
  float* dst;
  int chb;
  if (ot * 16 < split) { dst = dlo; chb = ot * 16; }
  else                 { dst = dhi; chb = ot * 16 - split; }
  // channel c = chb + v + hi*8, chb 16-aligned => head h = v, hidden = chb/8+hi
  const int hd = (chb >> 3) + hi;

  // groups: {x=0}, {1..4}, {5..10}, {11..14}, {15}; e0 map y = grp+4
#pragma unroll
  for (int grp = 0; grp < 5; ++grp) {
    const int xb = (grp == 0) ? 0 : (grp == 1) ? 1 : (grp == 2) ? 5
                 : (grp == 3) ? 11 : 15;
    const int xe = (grp == 0) ? 1 : (grp == 1) ? 5 : (grp == 2) ? 11
                 : (grp == 3) ? 15 : 16;
    const int e0max = (grp == 1) ? 1 : (grp == 2) ? 7 : (grp == 3) ? 13
                    : (grp == 4) ? 15 : -1;

    v2f ag[8], ae[8];
    {
      int abase = wmvBase + ((grp * OT + ot) * 8) * 64 + laneoff;
#pragma unroll
      for (int ks = 0; ks < 8; ++ks) ag[ks] = ldg_v2f(wsf + abase + ks * 64);
    }
    if (grp > 0) {  // e0 A-fragments, shared by the group's e0-dst blades
      int abase = wmvBase + (((grp + 4) * OT + ot) * 8) * 64 + laneoff;
#pragma unroll
      for (int ks = 0; ks < 8; ++ks) ae[ks] = ldg_v2f(wsf + abase + ks * 64);
    }

#pragma unroll 1
    for (int x = xb; x < xe; ++x) {
      v8f acc = {};
#pragma unroll
      for (int ks = 0; ks < 8; ++ks) {  // grade projection: K = 32
        v2f b = *(const v2f*)(mvlds + (x * 16 + t) * MV_STRIDE + ks * 4 + hi * 2);
        acc = wmma_f32(ag[ks], b, acc);
      }
      if (grp > 0 && x <= e0max) {  // e0 term: K = 32 from source blade
        int xs = (x == 1) ? 0 : (x == 15) ? 14 : x - 3;
#pragma unroll
        for (int ks = 0; ks < 8; ++ks) {
          v2f b = *(const v2f*)(mvlds + (xs * 16 + t) * MV_STRIDE + ks * 4 + hi * 2);
          acc = wmma_f32(ae[ks], b, acc);
        }
      }
      if (grp == 0) {  // scalar -> mv (blade 0 only): K = 128
#pragma unroll 8
        for (int ks = 0; ks < 32; ++ks) {
          v2f a = ldg_v2f(wsf + s2mvBase + (ot * 32 + ks) * 64 + laneoff);
          v2f b = *(const v2f*)(slds + t * S_STRIDE + ks * 4 + hi * 2);
          acc = wmma_f32(a, b, acc);
        }
      }
#pragma unroll
      for (int v = 0; v < 8; ++v)
        stg_f(dst + (((bb * 8 + v) * NTOK + n) * 8 + hd) * 16 + x, acc[v]);
    }
  }
}

// ---------------- one 16-channel scalar output tile ----------------
static __device__ void run_s_job(const float* __restrict__ wsf, int mv2sBase,
                                 int s2sBase, int ot,
                                 const float* __restrict__ mvlds,
                                 const float* __restrict__ slds,
                                 const float* __restrict__ bias,
                                 float* __restrict__ dlo,
                                 float* __restrict__ dhi, int split, int bb,
                                 int tile16, int lane) {
  const int t = lane & 15;
  const int hi = lane >> 4;
  const int laneoff = lane * 2;

  float* dst;
  int chb;
  if (ot * 16 < split) { dst = dlo; chb = ot * 16; }
  else                 { dst = dhi; chb = ot * 16 - split; }
  const int hd = (chb >> 3) + hi;

  v8f acc = {};
#pragma unroll
  for (int ks = 0; ks < 8; ++ks) {  // mv blade-0 -> scalar: K = 32
    v2f a = ldg_v2f(wsf + mv2sBase + (ot * 8 + ks) * 64 + laneoff);
    v2f b = *(const v2f*)(mvlds + t * MV_STRIDE + ks * 4 + hi * 2);
    acc = wmma_f32(a, b, acc);
  }
#pragma unroll 8
  for (int ks = 0; ks < 32; ++ks) {  // scalar -> scalar: K = 128
    v2f a = ldg_v2f(wsf + s2sBase + (ot * 32 + ks) * 64 + laneoff);
    v2f b = *(const v2f*)(slds + t * S_STRIDE + ks * 4 + hi * 2);
    acc = wmma_f32(a, b, acc);
  }
  int n = tile16 + t;
#pragma unroll
  for (int v = 0; v < 8; ++v) {
    float bv = ldg_f(bias + ot * 16 + v + hi * 8);
    stg_f(dst + ((bb * 8 + v) * NTOK + n) * 32 + hd, acc[v] + bv);
  }
}

// ---------------- main kernel: one WG per 16-token tile ----------------
__global__ __launch_bounds__(256) void crossattn_qkv_kernel(
    const float* __restrict__ inputs, const float* __restrict__ scalars,
    const int* __restrict__ qidx, const int* __restrict__ kvidx,
    const float* __restrict__ qbs, const float* __restrict__ kvbs,
    const float* __restrict__ wsf, float* __restrict__ out) {
  __shared__ float mvbuf[16 * 16 * MV_STRIDE];  // 36 KB
  __shared__ float sbuf[16 * S_STRIDE];         // 8.25 KB

  const int bb = blockIdx.x >> 8;      // 256 tiles per batch
  const int tile16 = (blockIdx.x & 255) * 16;
  const int tid = threadIdx.x;
  // readfirstlane => scalar job dispatch (s_cmp/s_cbranch, SGPR weight bases)
  const int wave = __builtin_amdgcn_readfirstlane(tid >> 5);
  const int lane = tid & 31;

  // ---- phase 1: q_idx-gathered inputs -> q_mv, v_mv_queries, q_s, v_s_queries
  stage_tile(inputs, scalars, qidx, bb, tile16, tid, mvbuf, sbuf);
  __syncthreads();
  for (int job = wave; job < 40; job += 8) {
    if (job < 4)
      run_mv_job(wsf, WQ_MV, WQ_S2MV, 4, job, mvbuf, sbuf, out + O_QMV,
                 out + O_QMV, 64, bb, tile16, lane);
    else if (job < 8)  // kv weights, ot 4..7 => v-half channels 64..127
      run_mv_job(wsf, WKV_MV, WKV_S2MV, 8, job, mvbuf, sbuf, out + O_VMVQ,
                 out + O_VMVQ, 64, bb, tile16, lane);
    else if (job < 24)
      run_s_job(wsf, WQ_MV2S, WQ_S2S, job - 8, mvbuf, sbuf, qbs, out + O_QS,
                out + O_QS, 256, bb, tile16, lane);
    else  // kv scalar weights, ot 16..31 => v_s channels 256..511
      run_s_job(wsf, WKV_MV2S, WKV_S2S, 16 + (job - 24), mvbuf, sbuf, kvbs,
                out + O_VSQ, out + O_VSQ, 256, bb, tile16, lane);
  }
  __syncthreads();

  // ---- phase 2: kv_idx-gathered inputs -> k_mv, v_mv, k_s, v_s
  stage_tile(inputs, scalars, kvidx, bb, tile16, tid, mvbuf, sbuf);
  __syncthreads();
  for (int job = wave; job < 40; job += 8) {
    if (job < 8)
      run_mv_job(wsf, WKV_MV, WKV_S2MV, 8, job, mvbuf, sbuf, out + O_KMV,
                 out + O_VMV, 64, bb, tile16, lane);
    else
      run_s_job(wsf, WKV_MV2S, WKV_S2S, job - 8, mvbuf, sbuf, kvbs, out + O_KS,
                out + O_VS, 256, bb, tile16, lane);
  }
}

extern "C" void kernel_launch(void* const* d_in, const int* in_sizes, int n_in,
                              void* d_out, int out_size, void* d_ws,
                              size_t ws_size, hipStream_t stream) {
  const float* inputs  = (const float*)d_in[0];
  const float* scalars = (const float*)d_in[1];
  const float* qwmv    = (const float*)d_in[2];
  const float* qws2mv  = (const float*)d_in[3];
  const float* qwmv2s  = (const float*)d_in[4];
  const float* qws2s   = (const float*)d_in[5];
  const float* qbs     = (const float*)d_in[6];
  const float* kvwmv   = (const float*)d_in[7];
  const float* kvws2mv = (const float*)d_in[8];
  const float* kvwmv2s = (const float*)d_in[9];
  const float* kvws2s  = (const float*)d_in[10];
  const float* kvbs    = (const float*)d_in[11];
  const int* qidx      = (const int*)d_in[12];
  const int* kvidx     = (const int*)d_in[13];
  float* out = (float*)d_out;
  float* wsf = (float*)d_ws;

  (void)in_sizes; (void)n_in; (void)out_size; (void)ws_size;

  // 1) repack all weight tensors into WMMA A-fragment layout (811 KB in ws)
  repack_kernel<<<dim3(128, 8), 256, 0, stream>>>(
      qwmv, kvwmv, qws2mv, kvws2mv, qwmv2s, kvwmv2s, qws2s, kvws2s, wsf);

  // 2) main compute: 2 batches * 256 token tiles
  crossattn_qkv_kernel<<<NB * 256, 256, 0, stream>>>(
      inputs, scalars, qidx, kvidx, qbs, kvbs, wsf, out);
}